// vector_contract_48412871360660
// MI455X (gfx1250) — compile-verified
//
#include <hip/hip_runtime.h>
#include <stdint.h>

// Problem constants (fixed by the reference harness).
static constexpr int Mdim = 65536;
static constexpr int Ndim = 64;
#define NNZ_CHUNK 64   // nonzeros staged per workgroup (1e6 % 64 == 0)

// Relaxed agent-scope f32 atomic add -> lowers to global_atomic_add_f32
// (confirmed by instruction counts: no CAS-loop blowup).
__device__ __forceinline__ void atom_add_f32(float* p, float v) {
  __hip_atomic_fetch_add(p, v, __ATOMIC_RELAXED, __HIP_MEMORY_SCOPE_AGENT);
}

// Low 32 bits of a flat pointer into __shared__ == wave-relative LDS byte
// address (ISA: LDS aperture uses addr[31:0]); this is what the async-to-LDS
// VDST operand wants.
__device__ __forceinline__ uint32_t lds_u32(const void* p) {
  return (uint32_t)(uintptr_t)p;
}

// One fused grid covering all three COO streams.
//   segment 0: c_tilde^T  (rows/cols pre-swapped at launch), real part
//   segment 1: au0,                                          real part
//   segment 2: au1,                                          imag part
// Real part : out_re += v*Bre[c], out_im += v*Bim[c]
// Imag part : out_re -= v*Bim[c], out_im += v*Bre[c]
// Expressed branchlessly: pre += (sgn*v)*Bfirst[c], pim += v*Bsecond[c]
// with (Bfirst,Bsecond,sgn) = (Bre,Bim,+1) or (Bim,Bre,-1).
__global__ __launch_bounds__(256) void spmm_scatter_fused(
    const int* __restrict__ rows0, const int* __restrict__ cols0,
    const float* __restrict__ vals0, int nb0, int nnz0,
    const int* __restrict__ rows1, const int* __restrict__ cols1,
    const float* __restrict__ vals1, int nb1, int nnz1,
    const int* __restrict__ rows2, const int* __restrict__ cols2,
    const float* __restrict__ vals2, int nnz2,
    const float* __restrict__ Bre, const float* __restrict__ Bim,
    float* __restrict__ out_re, float* __restrict__ out_im) {
  __shared__ int   sh_rows[NNZ_CHUNK];
  __shared__ int   sh_cols[NNZ_CHUNK];
  __shared__ float sh_vals[NNZ_CHUNK];

  // Block-uniform segment select.
  const int b = blockIdx.x;
  const int* rows;
  const int* cols;
  const float* vals;
  int nnz, blk;
  const float* Bf;   // feeds out_re (scaled by sgn*v)
  const float* Bs;   // feeds out_im (scaled by v)
  float sgn;
  if (b < nb0) {
    rows = rows0; cols = cols0; vals = vals0; nnz = nnz0; blk = b;
    Bf = Bre; Bs = Bim; sgn = 1.0f;
  } else if (b < nb0 + nb1) {
    rows = rows1; cols = cols1; vals = vals1; nnz = nnz1; blk = b - nb0;
    Bf = Bre; Bs = Bim; sgn = 1.0f;
  } else {
    rows = rows2; cols = cols2; vals = vals2; nnz = nnz2; blk = b - nb0 - nb1;
    Bf = Bim; Bs = Bre; sgn = -1.0f;
  }

  const int tid  = threadIdx.x;
  const int base = blk * NNZ_CHUNK;
  const int nrem = nnz - base;

  if (nrem >= NNZ_CHUNK) {
    // gfx1250 async copy: stage 3 x 256B of COO metadata into LDS.
    // 48 lanes x global_load_async_to_lds_b128 (tracked by ASYNCcnt).
    // Metadata is streamed once -> non-temporal hint keeps L2 for the
    // reused B rows and the atomically-updated output.
    if (tid < 48) {
      const int sel  = tid >> 4;   // 0: rows, 1: cols, 2: vals
      const int lane = tid & 15;
      const char* gsrc;
      uint32_t    ldst;
      if (sel == 0)      { gsrc = (const char*)(rows + base); ldst = lds_u32(sh_rows); }
      else if (sel == 1) { gsrc = (const char*)(cols + base); ldst = lds_u32(sh_cols); }
      else               { gsrc = (const char*)(vals + base); ldst = lds_u32(sh_vals); }
      gsrc += lane * 16;
      ldst += (uint32_t)(lane * 16);
      asm volatile("global_load_async_to_lds_b128 %0, %1, off th:TH_LOAD_NT"
                   :: "v"(ldst), "v"(gsrc)
                   : "memory");
    }
    asm volatile("s_wait_asynccnt 0" ::: "memory");
    __syncthreads();
  } else {
    // Tail fallback (never taken for NNZ % NNZ_CHUNK == 0, kept for safety).
    // Zero-pad so the processing loop below needs no bounds checks:
    // r=c=0, v=0 entries are numerical no-ops (x += +/-0.0f).
    if (tid < NNZ_CHUNK) {
      if (base + tid < nnz) {
        sh_rows[tid] = rows[base + tid];
        sh_cols[tid] = cols[base + tid];
        sh_vals[tid] = vals[base + tid];
      } else {
        sh_rows[tid] = 0; sh_cols[tid] = 0; sh_vals[tid] = 0.0f;
      }
    }
    __syncthreads();
  }

  // 16 groups of 16 lanes; each lane owns 4 channels (float4 = b128).
  // Branch-free, fully unrolled: all 8 B-row loads can issue before the
  // first atomic, maximizing memory-level parallelism.
  const int grp    = tid >> 4;
  const int lane16 = tid & 15;

#pragma unroll
  for (int t = 0; t < NNZ_CHUNK / 16; ++t) {
    const int k = grp * (NNZ_CHUNK / 16) + t;
    const int   r = sh_rows[k];
    const int   c = sh_cols[k];
    const float v = sh_vals[k];
    const float cf = sgn * v;

    const float4 bf = ((const float4*)(Bf + (size_t)c * Ndim))[lane16];
    const float4 bs = ((const float4*)(Bs + (size_t)c * Ndim))[lane16];

    float* pre = out_re + (size_t)r * Ndim + lane16 * 4;
    float* pim = out_im + (size_t)r * Ndim + lane16 * 4;

    atom_add_f32(pre + 0, cf * bf.x);
    atom_add_f32(pre + 1, cf * bf.y);
    atom_add_f32(pre + 2, cf * bf.z);
    atom_add_f32(pre + 3, cf * bf.w);
    atom_add_f32(pim + 0, v * bs.x);
    atom_add_f32(pim + 1, v * bs.y);
    atom_add_f32(pim + 2, v * bs.z);
    atom_add_f32(pim + 3, v * bs.w);
  }
}

extern "C" void kernel_launch(void* const* d_in, const int* in_sizes, int n_in,
                              void* d_out, int out_size, void* d_ws, size_t ws_size,
                              hipStream_t stream) {
  const int*   ct_rows  = (const int*)d_in[0];
  const int*   ct_cols  = (const int*)d_in[1];
  const float* ct_vals  = (const float*)d_in[2];
  const int*   au0_rows = (const int*)d_in[3];
  const int*   au0_cols = (const int*)d_in[4];
  const float* au0_vals = (const float*)d_in[5];
  const int*   au1_rows = (const int*)d_in[6];
  const int*   au1_cols = (const int*)d_in[7];
  const float* au1_vals = (const float*)d_in[8];
  const float* ab_re    = (const float*)d_in[9];
  const float* ab_im    = (const float*)d_in[10];

  float* out_re = (float*)d_out;                       // [M,64]
  float* out_im = out_re + (size_t)Mdim * Ndim;        // [M,64]

  // d_out is poisoned once before timing; we must zero it every call.
  hipMemsetAsync(d_out, 0, (size_t)out_size * sizeof(float), stream);

  const int nnz_ct  = in_sizes[0];
  const int nnz_au0 = in_sizes[3];
  const int nnz_au1 = in_sizes[6];

  auto nblocks = [](int nnz) { return (nnz + NNZ_CHUNK - 1) / NNZ_CHUNK; };
  const int nb0 = nblocks(nnz_ct);
  const int nb1 = nblocks(nnz_au0);
  const int nb2 = nblocks(nnz_au1);

  // Single fused grid; c_tilde is used transposed -> swap row/col pointers.
  spmm_scatter_fused<<<nb0 + nb1 + nb2, 256, 0, stream>>>(
      ct_cols, ct_rows, ct_vals, nb0, nnz_ct,
      au0_rows, au0_cols, au0_vals, nb1, nnz_au0,
      au1_rows, au1_cols, au1_vals, nnz_au1,
      ab_re, ab_im, out_re, out_im);
}